// GATv2_64665027609336
// MI455X (gfx1250) — compile-verified
//
#include <hip/hip_runtime.h>
#include <math.h>

typedef __attribute__((ext_vector_type(2))) float v2f;
typedef __attribute__((ext_vector_type(8))) float v8f;

#define TPB 256
#define WAVES_PER_BLOCK 8   // 256 threads / wave32

// ---------------------------------------------------------------------------
// GEMM: Y[N,M] = X[N,64] @ W[64,M], fp32 via V_WMMA_F32_16X16X4_F32.
// Block (8 waves) stages W in LDS once; each wave owns one 16-row tile,
// caches its A fragments in registers, then sweeps all M/16 column tiles.
// M is a template constant so all strides are immediate offsets.
//
// A layout (16x4 f32): lanes 0-15 hold M=0..15, v0=K0,v1=K1; lanes 16-31 K2/K3.
// B layout (4x16 f32): lanes 0-15 N=0..15 rows K0(v0)/K1(v1); lanes 16-31 K2/K3.
// C/D (16x16 f32): lane 0-15 N=lane, VGPR v -> M=v; lanes 16-31 -> M=v+8.
// ---------------------------------------------------------------------------
template <int M>
__global__ __launch_bounds__(TPB) void gemm_wmma_f32(
    const float* __restrict__ X, const float* __restrict__ W,
    float* __restrict__ Y, int Nrows) {
  __shared__ __align__(16) float lw[64 * M];

  // Cooperative stage of W[64,M] into LDS (float4, fully coalesced).
  constexpr int TOTAL4 = 64 * M / 4;
  for (int i = threadIdx.x; i < TOTAL4; i += TPB)
    ((float4*)lw)[i] = ((const float4*)W)[i];
  __syncthreads();

  const int rowTile = blockIdx.x * WAVES_PER_BLOCK + (threadIdx.x >> 5);
  if (rowTile >= (Nrows >> 4)) return;        // wave-uniform: EXEC stays all-1s
  const int row0 = rowTile << 4;
  const int lane = threadIdx.x & 31;
  const int r    = lane & 15;                 // row (A) / col (B,C) within tile
  const int half = lane >> 4;                 // K sub-pair selector

  // Cache the 16x64 A tile: 16 k-steps x v2f per lane (32 VGPRs).
  v2f a[16];
  const float* xrow = X + (size_t)(row0 + r) * 64;
#pragma unroll
  for (int ks = 0; ks < 16; ++ks) {
    const int k = 4 * ks + 2 * half;
    a[ks].x = xrow[k];
    a[ks].y = xrow[k + 1];
  }

#pragma unroll
  for (int tn = 0; tn < M / 16; ++tn) {
    const int col0 = tn << 4;
    v8f c = {};
#pragma unroll
    for (int ks = 0; ks < 16; ++ks) {
      const int k = 4 * ks + 2 * half;
      v2f b;
      b.x = lw[k * M + col0 + r];             // immediate-offset ds_load
      b.y = lw[(k + 1) * M + col0 + r];
      c = __builtin_amdgcn_wmma_f32_16x16x4_f32(false, a[ks], false, b,
                                                (short)0, c, false, false);
    }
#pragma unroll
    for (int v = 0; v < 8; ++v)
      Y[(size_t)(row0 + v + 8 * half) * M + col0 + r] = c[v];
  }
}

// ---------------------------------------------------------------------------
// Edge helpers
// ---------------------------------------------------------------------------
__device__ __forceinline__ void edge_sd(const int* __restrict__ ei, int e, int E,
                                        int& s, int& d) {
  if (e < E) { s = ei[e]; d = ei[E + e]; }
  else       { s = e - E; d = s; }           // appended self-loops
}

// float atomic-max via signed/unsigned integer monotonicity trick.
// Init bytes must be 0xFF; every slot receives >=1 update (self-loops).
__device__ __forceinline__ void atomic_max_float(float* addr, float val) {
  if (val >= 0.0f) atomicMax((int*)addr, __float_as_int(val));
  else             atomicMin((unsigned int*)addr, __float_as_uint(val));
}

// Pass 1: score[e,h] = sum_c att[h,c]*leakyrelu(xl[src,h,c]+xr[dst,h,c], 0.2)
//         m[dst,h]   = max(score)
template <int C>
__global__ __launch_bounds__(TPB) void edge_score(
    const float* __restrict__ xl, const float* __restrict__ xr,
    const int* __restrict__ ei, const float* __restrict__ att,
    float* __restrict__ score, float* __restrict__ mbuf, int E, int ET) {
  const int tid = blockIdx.x * blockDim.x + threadIdx.x;
  if (tid >= ET * 4) return;
  const int e = tid >> 2, h = tid & 3;
  int s, d; edge_sd(ei, e, E, s, d);
  const int ldf = 4 * C;
  const float4* pl = (const float4*)(xl + (size_t)s * ldf + h * C);
  const float4* pr = (const float4*)(xr + (size_t)d * ldf + h * C);
  const float4* pa = (const float4*)(att + h * C);
  float acc = 0.0f;
#pragma unroll
  for (int i = 0; i < C / 4; ++i) {
    float4 a = pl[i], b = pr[i], w = pa[i];
    float vx = a.x + b.x, vy = a.y + b.y, vz = a.z + b.z, vw = a.w + b.w;
    vx = vx > 0.0f ? vx : 0.2f * vx;
    vy = vy > 0.0f ? vy : 0.2f * vy;
    vz = vz > 0.0f ? vz : 0.2f * vz;
    vw = vw > 0.0f ? vw : 0.2f * vw;
    acc += w.x * vx + w.y * vy + w.z * vz + w.w * vw;
  }
  score[(size_t)e * 4 + h] = acc;
  atomic_max_float(&mbuf[(size_t)d * 4 + h], acc);
}

// Pass 2: ex = exp(score - m[dst]); denom[dst,h] += ex (score overwritten)
__global__ __launch_bounds__(TPB) void edge_exp(
    float* __restrict__ score, const float* __restrict__ mbuf,
    float* __restrict__ denom, const int* __restrict__ ei, int E, int ET) {
  const int tid = blockIdx.x * blockDim.x + threadIdx.x;
  if (tid >= ET * 4) return;
  const int e = tid >> 2, h = tid & 3;
  int s, d; edge_sd(ei, e, E, s, d);
  const float ex = expf(score[(size_t)e * 4 + h] - mbuf[(size_t)d * 4 + h]);
  score[(size_t)e * 4 + h] = ex;
  atomicAdd(&denom[(size_t)d * 4 + h], ex);
}

// Pass 3: agg[dst,h,:] += (ex/denom[dst,h]) * xl[src,h,:]
template <int C>
__global__ __launch_bounds__(TPB) void edge_agg(
    const float* __restrict__ xl, const float* __restrict__ score,
    const float* __restrict__ denom, const int* __restrict__ ei,
    float* __restrict__ agg, int E, int ET) {
  const int tid = blockIdx.x * blockDim.x + threadIdx.x;
  if (tid >= ET * 4) return;
  const int e = tid >> 2, h = tid & 3;
  int s, d; edge_sd(ei, e, E, s, d);
  const float alpha = score[(size_t)e * 4 + h] / denom[(size_t)d * 4 + h];
  const int ldf = 4 * C;
  const float4* srcp = (const float4*)(xl + (size_t)s * ldf + h * C);
  float* dstp = agg + (size_t)d * ldf + h * C;
#pragma unroll
  for (int i = 0; i < C / 4; ++i) {
    float4 v = srcp[i];
    atomicAdd(&dstp[4 * i + 0], alpha * v.x);
    atomicAdd(&dstp[4 * i + 1], alpha * v.y);
    atomicAdd(&dstp[4 * i + 2], alpha * v.z);
    atomicAdd(&dstp[4 * i + 3], alpha * v.w);
  }
}

// Node epilogue for layers 1-2: in-place feat = elu(feat + bias)
__global__ __launch_bounds__(TPB) void bias_elu(
    float* __restrict__ feat, const float* __restrict__ bias, int total, int M) {
  const int tid = blockIdx.x * blockDim.x + threadIdx.x;
  if (tid >= total) return;
  const float v = feat[tid] + bias[tid & (M - 1)];   // M = 64 (power of two)
  feat[tid] = v > 0.0f ? v : expm1f(v);
}

// Final layer: mean over H=4 heads (C=32) + bias -> logits [N,32]
__global__ __launch_bounds__(TPB) void final_mean(
    const float* __restrict__ agg, const float* __restrict__ bias,
    float* __restrict__ out, int N) {
  const int tid = blockIdx.x * blockDim.x + threadIdx.x;
  if (tid >= N * 32) return;
  const int n = tid >> 5, c = tid & 31;
  const size_t b = (size_t)n * 128 + c;
  const float s = 0.25f * (agg[b] + agg[b + 32] + agg[b + 64] + agg[b + 96]);
  out[tid] = s + bias[c];
}

// ---------------------------------------------------------------------------
extern "C" void kernel_launch(void* const* d_in, const int* in_sizes, int n_in,
                              void* d_out, int out_size, void* d_ws, size_t ws_size,
                              hipStream_t stream) {
  (void)in_sizes; (void)n_in; (void)out_size; (void)ws_size;
  const float* x   = (const float*)d_in[0];
  const int*   ei  = (const int*)d_in[1];
  const float* W1l = (const float*)d_in[2];
  const float* W1r = (const float*)d_in[3];
  const float* a1  = (const float*)d_in[4];
  const float* b1  = (const float*)d_in[5];
  const float* W2l = (const float*)d_in[6];
  const float* W2r = (const float*)d_in[7];
  const float* a2  = (const float*)d_in[8];
  const float* b2  = (const float*)d_in[9];
  const float* W3l = (const float*)d_in[10];
  const float* W3r = (const float*)d_in[11];
  const float* a3  = (const float*)d_in[12];
  const float* b3  = (const float*)d_in[13];

  const int N = 100000, E = 800000, ET = E + N;  // self-loops appended

  float* ws = (float*)d_ws;   // flat fp32 workspace; all offsets 16B aligned
  float* xl = ws;                         // N*128 (max M)
  float* xr = xl + (size_t)N * 128;       // N*128
  float* sc = xr + (size_t)N * 128;       // ET*4
  float* mb = sc + (size_t)ET * 4;        // N*4
  float* dn = mb + (size_t)N * 4;         // N*4
  float* h1 = dn + (size_t)N * 4;         // N*64
  float* h2 = h1 + (size_t)N * 64;        // N*64
  float* ag = h2 + (size_t)N * 64;        // N*128 (layer-3 pre-mean agg)

  const int rowTiles = N / 16;                                   // 6250
  const int gblocks  = (rowTiles + WAVES_PER_BLOCK - 1) / WAVES_PER_BLOCK;

  auto layer = [&](const float* in, const float* Wl, const float* Wr,
                   const float* att, const float* bias, float* out,
                   int C, bool final_layer) {
    const int M = 4 * C;
    if (M == 64) {
      gemm_wmma_f32<64><<<gblocks, TPB, 0, stream>>>(in, Wl, xl, N);
      gemm_wmma_f32<64><<<gblocks, TPB, 0, stream>>>(in, Wr, xr, N);
    } else {
      gemm_wmma_f32<128><<<gblocks, TPB, 0, stream>>>(in, Wl, xl, N);
      gemm_wmma_f32<128><<<gblocks, TPB, 0, stream>>>(in, Wr, xr, N);
    }

    float* aggbuf = final_layer ? ag : out;
    hipMemsetAsync(mb, 0xFF, (size_t)N * 4 * sizeof(float), stream); // init for max-trick
    hipMemsetAsync(dn, 0x00, (size_t)N * 4 * sizeof(float), stream);
    hipMemsetAsync(aggbuf, 0x00, (size_t)N * M * sizeof(float), stream);

    const int eThreads = ET * 4;
    const int eBlocks = (eThreads + TPB - 1) / TPB;
    if (C == 16) edge_score<16><<<eBlocks, TPB, 0, stream>>>(xl, xr, ei, att, sc, mb, E, ET);
    else         edge_score<32><<<eBlocks, TPB, 0, stream>>>(xl, xr, ei, att, sc, mb, E, ET);
    edge_exp<<<eBlocks, TPB, 0, stream>>>(sc, mb, dn, ei, E, ET);
    if (C == 16) edge_agg<16><<<eBlocks, TPB, 0, stream>>>(xl, sc, dn, ei, aggbuf, E, ET);
    else         edge_agg<32><<<eBlocks, TPB, 0, stream>>>(xl, sc, dn, ei, aggbuf, E, ET);

    if (!final_layer) {
      const int t = N * M;
      bias_elu<<<(t + TPB - 1) / TPB, TPB, 0, stream>>>(out, bias, t, M);
    } else {
      const int t = N * 32;
      final_mean<<<(t + TPB - 1) / TPB, TPB, 0, stream>>>(ag, bias, (float*)d_out, N);
    }
  };

  layer(x,  W1l, W1r, a1, b1, h1, 16, false);
  layer(h1, W2l, W2r, a2, b2, h2, 16, false);
  layer(h2, W3l, W3r, a3, b3, (float*)d_out, 32, true);
}